// BarycentricInterpolator_63720134803868
// MI455X (gfx1250) — compile-verified
//
#include <hip/hip_runtime.h>
#include <stdint.h>

// Barycentric interpolation: out[b,m] = sum_k f[b, tri[m,k]] * w[m,k]
// B=128, N=10000, M=500000.  Memory-bound (~273 MB traffic, ~0.38 GFLOP):
// the 256 MB output write dominates.  Strategy:
//  - one thread per target point, loop over the 128 batch rows (coalesced stores)
//  - stage idx/weights tiles into LDS via gfx1250 global_load_async_to_lds_b32
//    (ASYNCcnt path), synced with s_wait_asynccnt + barrier
//  - non-temporal output stores so the 256 MB stream does not evict the
//    5 MB f_values working set from the 192 MB L2 / 64 KB WGP$.

namespace {
constexpr int kB   = 128;     // batch of scalar fields
constexpr int kN   = 10000;   // base mesh vertices
constexpr int kM   = 500000;  // target points
constexpr int kTPB = 256;     // threads per block == points per tile
}

__global__ __launch_bounds__(kTPB) void bary_interp_kernel(
    const float* __restrict__ f,    // (B, N)
    const int*   __restrict__ tri,  // (M, 3)
    const float* __restrict__ bw,   // (M, 3)
    float* __restrict__ out)        // (B, M)
{
  __shared__ int   s_idx[kTPB * 3];
  __shared__ float s_w[kTPB * 3];

  const int t = threadIdx.x;
  const long long tileBase = (long long)blockIdx.x * kTPB;  // first point of tile
  const long long dwBase   = tileBase * 3;                  // first dword in (M,3)
  const long long dwTotal  = (long long)kM * 3;

  // LDS byte offsets (low 32 bits of a generic pointer to LDS == LDS address).
  const unsigned ldsI = (unsigned)(uintptr_t)&s_idx[0];
  const unsigned ldsW = (unsigned)(uintptr_t)&s_w[0];

  // ---- async-stage this tile's indices & weights into LDS (coalesced) ----
#pragma unroll
  for (int j = 0; j < 3; ++j) {
    const int e = t + j * kTPB;                  // dword index within tile
    const bool in = (dwBase + e) < dwTotal;      // clamp tail tile in-bounds
    const int*   gi = in ? (tri + dwBase + e) : tri;
    const float* gw = in ? (bw  + dwBase + e) : bw;
    const unsigned li = ldsI + 4u * (unsigned)e;
    const unsigned lw = ldsW + 4u * (unsigned)e;
    asm volatile("global_load_async_to_lds_b32 %0, %1, off"
                 :: "v"(li), "v"(gi) : "memory");
    asm volatile("global_load_async_to_lds_b32 %0, %1, off"
                 :: "v"(lw), "v"(gw) : "memory");
  }
  asm volatile("s_wait_asynccnt 0" ::: "memory");
  __syncthreads();

  const long long m = tileBase + t;
  if (m >= kM) return;

  const int   i0 = s_idx[3 * t + 0];
  const int   i1 = s_idx[3 * t + 1];
  const int   i2 = s_idx[3 * t + 2];
  const float w0 = s_w[3 * t + 0];
  const float w1 = s_w[3 * t + 1];
  const float w2 = s_w[3 * t + 2];

  const float* frow = f;        // row b of f_values (advances by N)
  float*       orow = out + m;  // out[b*M + m]     (advances by M)
#pragma unroll 4
  for (int b = 0; b < kB; ++b) {
    const float v = fmaf(frow[i0], w0, fmaf(frow[i1], w1, frow[i2] * w2));
    __builtin_nontemporal_store(v, orow);  // keep 256 MB stream out of L2
    frow += kN;
    orow += kM;
  }
}

extern "C" void kernel_launch(void* const* d_in, const int* in_sizes, int n_in,
                              void* d_out, int out_size, void* d_ws, size_t ws_size,
                              hipStream_t stream) {
  const float* f   = (const float*)d_in[0];  // (128, 10000) f32
  const int*   tri = (const int*)  d_in[1];  // (500000, 3) i32
  const float* bw  = (const float*)d_in[2];  // (500000, 3) f32
  float*       out = (float*)d_out;          // (128, 500000) f32

  const int tiles = (kM + kTPB - 1) / kTPB;  // 1954 workgroups of 8 wave32s
  hipLaunchKernelGGL(bary_interp_kernel, dim3(tiles), dim3(kTPB), 0, stream,
                     f, tri, bw, out);

  (void)in_sizes; (void)n_in; (void)out_size; (void)d_ws; (void)ws_size;
}